// ChamferDistanceLoss_16071767621914
// MI455X (gfx1250) — compile-verified
//
#include <hip/hip_runtime.h>
#include <hip/hip_bf16.h>

typedef __attribute__((ext_vector_type(2))) float v2f;
typedef __attribute__((ext_vector_type(8))) float v8f;

#define B_  8
#define N_  4096
#define M_  4096
#define TILES_ (M_ / 16)
#define BIGF 1e30f

__device__ __forceinline__ int imin(int a, int b) { return a < b ? a : b; }
__device__ __forceinline__ int imax(int a, int b) { return a > b ? a : b; }

// ---------------------------------------------------------------------------
// Kernel 0: init the cham_y min workspace (d_ws is poisoned by the harness).
// ---------------------------------------------------------------------------
__global__ __launch_bounds__(256) void chamfer_init_miny(float* __restrict__ gminy) {
    int i = blockIdx.x * blockDim.x + threadIdx.x;
    if (i < B_ * M_) gminy[i] = BIGF;
}

// ---------------------------------------------------------------------------
// Kernel 1: tiled pairwise d2 via V_WMMA_F32_16X16X4_F32.
//   grid = B * (N/128) blocks, 128 threads (4 waves). Each wave owns TWO
//   16-row fg tiles (rows rowBase..rowBase+31); one B-operand build feeds
//   2 WMMAs. prj tile loads are software-pipelined one iteration ahead.
//   A-row = (-2x, -2y, -2z, ||a||^2), B-col = (x', y', z', 1), C = ||b||^2
//   => D[r][c] = ||a_r||^2 + ||b_c||^2 - 2 a_r.b_c  in one WMMA per tile.
//   Clamp + all mins run on the int bit patterns (max_i32 with 0 == clamp
//   to +0.0 for non-NaN floats; non-negative floats order as signed ints),
//   so the whole reduction is canonicalize-free v_max_i32/v_min3_i32.
// ---------------------------------------------------------------------------
__global__ __launch_bounds__(128) void chamfer_tiles(
    const float* __restrict__ fg, const float* __restrict__ prj,
    int* __restrict__ gminx, float* __restrict__ gminy) {

    __shared__ int lds_miny[M_];     // 16 KB of the 320 KB WGP LDS (float bits)

    const int tid  = threadIdx.x;
    const int lane = tid & 31;
    const int wave = tid >> 5;       // 0..3
    const int lr   = lane & 15;      // row/col index within 16-wide tile
    const int half = lane >> 4;      // K-split half (K=0,1 vs K=2,3)

    const int b       = blockIdx.x >> 5;   // 32 row-groups of 128 per batch
    const int rg      = blockIdx.x & 31;
    const int rowBase = rg * 128 + wave * 32;   // this wave: rows [rowBase, rowBase+32)

    const int BIGBITS = __float_as_int(BIGF);
    for (int i = tid; i < M_; i += 128) lds_miny[i] = BIGBITS;
    __syncthreads();

    // ---- two A operands (loaded once per wave) --------------------------
    v2f A0, A1;
    {
        const float* fp = fg + ((size_t)b * N_ + (size_t)(rowBase + lr)) * 3;
        float ax = fp[0], ay = fp[1], az = fp[2];
        float na = fmaf(ax, ax, fmaf(ay, ay, az * az));
        A0.x = half ? (-2.0f * az) : (-2.0f * ax);
        A0.y = half ? na           : (-2.0f * ay);
    }
    {
        const float* fp = fg + ((size_t)b * N_ + (size_t)(rowBase + 16 + lr)) * 3;
        float ax = fp[0], ay = fp[1], az = fp[2];
        float na = fmaf(ax, ax, fmaf(ay, ay, az * az));
        A1.x = half ? (-2.0f * az) : (-2.0f * ax);
        A1.y = half ? na           : (-2.0f * ay);
    }

    int r0[8], r1[8];                // running cham_x mins (float bits)
    #pragma unroll
    for (int v = 0; v < 8; ++v) { r0[v] = BIGBITS; r1[v] = BIGBITS; }

    const float* pbase = prj + (size_t)b * M_ * 3;

    // ---- software pipeline: preload tile 0 ------------------------------
    float px, py, pz;
    {
        const float* pp = pbase + (size_t)lr * 3;
        px = pp[0]; py = pp[1]; pz = pp[2];
    }

    for (int t = 0; t < TILES_; ++t) {
        // ---- issue next tile's load (uniform scalar index, SALU min) ----
        int nt = t + 1 < TILES_ ? t + 1 : TILES_ - 1;
        const float* pp = pbase + (size_t)(nt * 16 + lr) * 3;
        __builtin_prefetch(pp + 16 * 3 * 8, 0, 1);   // ~8 tiles further ahead
        float nx = pp[0], ny = pp[1], nz = pp[2];

        // ---- B operand + per-column norm from the pipelined registers ---
        float nb = fmaf(px, px, fmaf(py, py, pz * pz));
        v2f Bm;
        Bm.x = half ? pz   : px;
        Bm.y = half ? 1.0f : py;
        v8f C0, C1;
        #pragma unroll
        for (int v = 0; v < 8; ++v) { C0[v] = nb; C1[v] = nb; }

        v8f D0 = __builtin_amdgcn_wmma_f32_16x16x4_f32(
            false, A0, false, Bm, (short)0, C0, false, false);
        v8f D1 = __builtin_amdgcn_wmma_f32_16x16x4_f32(
            false, A1, false, Bm, (short)0, C1, false, false);

        int cm = BIGBITS;
        #pragma unroll
        for (int v = 0; v < 8; ++v) {
            int d0 = imax(__float_as_int(D0[v]), 0);   // clamp to +0.0, int-space
            int d1 = imax(__float_as_int(D1[v]), 0);
            r0[v] = imin(r0[v], d0);                   // v_min_i32, no canon
            r1[v] = imin(r1[v], d1);
            cm    = imin(cm, imin(d0, d1));            // v_min3_i32 tree
        }
        // combine halves: min over all 32 fg rows per column (swizzle or-16: lanes 0-15 read lane+16)
        cm = imin(cm, __builtin_amdgcn_ds_swizzle(cm, 0x021F));
        atomicMin(&lds_miny[t * 16 + lr], cm);         // ds_min_i32

        px = nx; py = ny; pz = nz;                     // rotate pipeline
    }

    // ---- cham_x: reduce running mins across the 16 lanes of each half ----
    #pragma unroll
    for (int mask = 1; mask <= 8; mask <<= 1) {
        #pragma unroll
        for (int v = 0; v < 8; ++v) {
            r0[v] = imin(r0[v], __shfl_xor(r0[v], mask));
            r1[v] = imin(r1[v], __shfl_xor(r1[v], mask));
        }
    }
    if (lr == 0) {   // lanes 0 and 16 each own 8 rows of each tile; no atomics
        int* op0 = gminx + (size_t)b * N_ + (size_t)(rowBase + half * 8);
        int* op1 = op0 + 16;
        #pragma unroll
        for (int v = 0; v < 8; ++v) { op0[v] = r0[v]; op1[v] = r1[v]; }
    }

    // ---- cham_y: flush workgroup-combined LDS mins once -----------------
    __syncthreads();
    for (int i = tid; i < M_; i += 128) {
        atomicMin((int*)&gminy[(size_t)b * M_ + i], lds_miny[i]);
    }
}

// ---------------------------------------------------------------------------
// Kernel 2: masked means + batch mean -> scalar
// ---------------------------------------------------------------------------
__global__ __launch_bounds__(256) void chamfer_finalize(
    const float* __restrict__ gminx, const float* __restrict__ gminy,
    const int* __restrict__ xlen, float* __restrict__ out) {

    __shared__ float red[256];
    const int tid = threadIdx.x;
    float total = 0.0f;

    for (int b = 0; b < B_; ++b) {
        const int L = xlen[b];
        float sx = 0.0f, sy = 0.0f;
        for (int i = tid; i < N_; i += 256)
            if (i < L) sx += gminx[(size_t)b * N_ + i];
        for (int i = tid; i < M_; i += 256)
            sy += gminy[(size_t)b * M_ + i];
        total += sx / (float)L + sy / (float)M_;
    }

    red[tid] = total;
    __syncthreads();
    for (int s = 128; s > 0; s >>= 1) {
        if (tid < s) red[tid] += red[tid + s];
        __syncthreads();
    }
    if (tid == 0) out[0] = red[0] / (float)B_;
}

// ---------------------------------------------------------------------------
extern "C" void kernel_launch(void* const* d_in, const int* in_sizes, int n_in,
                              void* d_out, int out_size, void* d_ws, size_t ws_size,
                              hipStream_t stream) {
    (void)in_sizes; (void)n_in; (void)out_size; (void)ws_size;

    const float* fg  = (const float*)d_in[0];   // [B,N,3] f32 (PAD-padded rows)
    const float* prj = (const float*)d_in[1];   // [B,M,3] f32
    const int*   xl  = (const int*)d_in[2];     // [B] int32
    float*       out = (float*)d_out;           // scalar f32

    float* gminx = (float*)d_ws;                // [B,N]  (fully overwritten, bit-stored)
    float* gminy = gminx + (size_t)B_ * N_;     // [B,M]  (atomic-min target)

    chamfer_init_miny<<<(B_ * M_ + 255) / 256, 256, 0, stream>>>(gminy);
    chamfer_tiles<<<B_ * (N_ / 128), 128, 0, stream>>>(fg, prj, (int*)gminx, gminy);
    chamfer_finalize<<<1, 256, 0, stream>>>(gminx, gminy, xl, out);
}